// CRF_21182778704919
// MI455X (gfx1250) — compile-verified
//
#include <hip/hip_runtime.h>
#include <hip/hip_bf16.h>

// ---------------------------------------------------------------------------
// Dense CRF, MI455X (gfx1250, wave32).
//  - Pairwise kernels recomputed on the fly (nothing N^2 ever hits memory).
//  - Feature Grams computed with V_WMMA_F32_16X16X4_F32 (exact f32 math).
//  - Probs + color features staged once per block into LDS (294 KB of the
//    320 KB/WGP CDNA5 LDS), streamed by all 8 waves.
// ---------------------------------------------------------------------------

typedef __attribute__((ext_vector_type(2))) float v2f;
typedef __attribute__((ext_vector_type(8))) float v8f;

#define Hh 96
#define Wd 96
#define Npix 9216          // 96*96
#define NTILES 576         // Npix / 16
#define WAVES_PER_BLK 8
#define BLK 256
#define NBLK_BLUR 72       // NTILES / WAVES_PER_BLK

__device__ __forceinline__ float inv_ss() { return 1.0f / 64.0f; } // 1/BIL_SS (==1/GAU_SS)
__device__ __forceinline__ float inv_cs() { return 5.0f; }         // 1/BIL_CS

// ---------------------------------------------------------------------------
// Init: probs0 = softmax(logits); color features = rgb/0.2 with |f|^2 in .w
// ---------------------------------------------------------------------------
__global__ void crf_init(const float* __restrict__ logits,
                         const float* __restrict__ ref,
                         float4* __restrict__ probs,
                         float4* __restrict__ cfeat) {
  int n = blockIdx.x * blockDim.x + threadIdx.x;
  if (n >= Npix) return;
  float l0 = logits[0 * Npix + n], l1 = logits[1 * Npix + n];
  float l2 = logits[2 * Npix + n], l3 = logits[3 * Npix + n];
  float m = fmaxf(fmaxf(l0, l1), fmaxf(l2, l3));
  float e0 = __expf(l0 - m), e1 = __expf(l1 - m);
  float e2 = __expf(l2 - m), e3 = __expf(l3 - m);
  float inv = 1.0f / (e0 + e1 + e2 + e3);
  probs[n] = make_float4(e0 * inv, e1 * inv, e2 * inv, e3 * inv);
  float r = ref[0 * Npix + n] * inv_cs();
  float g = ref[1 * Npix + n] * inv_cs();
  float b = ref[2 * Npix + n] * inv_cs();
  cfeat[n] = make_float4(r, g, b, r * r + g * g + b * b);
}

// ---------------------------------------------------------------------------
// Blur: comb[n,c] = sum_m [ exp(-.5 ds)*(0.8*exp(-.5 dc)+0.2) ] * p[m,c]
// One wave per 16-row n-tile. Per 16-col m-tile:
//   Ss = Xn^T Xm  (spatial Gram, K=2 padded to 4)   -> 1 WMMA
//   Sc = Cn^T Cm  (color  Gram, K=3 padded to 4)    -> 1 WMMA
//   d2 = sq[n]+sq[m]-2S ; w = exp ; acc[n][c] += w * p[m][c]
// A layout (ISA 7.12.2, 32-bit A 16x4): lane l -> row M=l%16,
//   VGPR{0,1} = K{0,1} (lanes 0-15) / K{2,3} (lanes 16-31).
// B assumed symmetric (col N=l%16, same K split). D: lane owns col N=l%16,
//   VGPR r -> row r + 8*(l>=16).
// ---------------------------------------------------------------------------
__global__ void __launch_bounds__(BLK) crf_blur(const float4* __restrict__ probs,
                                                const float4* __restrict__ cfeat,
                                                float4* __restrict__ comb) {
  extern __shared__ float4 smem[];
  float4* sC = smem;          // [Npix] color features (r,g,b,|f|^2)
  float4* sP = smem + Npix;   // [Npix] probabilities
  for (int i = threadIdx.x; i < Npix; i += BLK) {
    sC[i] = cfeat[i];
    sP[i] = probs[i];
  }
  __syncthreads();

  const int lane = threadIdx.x & 31;
  const int wave = threadIdx.x >> 5;
  const int tile = blockIdx.x * WAVES_PER_BLK + wave;
  const int nbase = tile * 16;
  const int col = lane & 15;
  const int half = lane >> 4;

  // A operands: features of row pixel nbase+col, K split across half-waves.
  int npix = nbase + col;
  float xi = (float)(npix / Wd) * inv_ss();
  float yj = (float)(npix % Wd) * inv_ss();
  float4 cn = sC[npix];
  v2f a_s, a_c;
  if (half == 0) { a_s = (v2f){xi, yj};   a_c = (v2f){cn.x, cn.y}; }
  else           { a_s = (v2f){0.f, 0.f}; a_c = (v2f){cn.z, 0.f};  }

  // Per-lane row norms for the 8 D rows this lane owns: n = nbase + r + 8*half
  float sqs_n[8], sqc_n[8];
#pragma unroll
  for (int r = 0; r < 8; ++r) {
    int nn = nbase + r + 8 * half;
    float xr = (float)(nn / Wd) * inv_ss();
    float yr = (float)(nn % Wd) * inv_ss();
    sqs_n[r] = xr * xr + yr * yr;
    sqc_n[r] = sC[nn].w;
  }

  float4 acc[8];
#pragma unroll
  for (int r = 0; r < 8; ++r) acc[r] = make_float4(0.f, 0.f, 0.f, 0.f);

  for (int mt = 0; mt < NTILES; ++mt) {
    int mpix = mt * 16 + col;
    float xm = (float)(mpix / Wd) * inv_ss();
    float ym = (float)(mpix % Wd) * inv_ss();
    float4 cm = sC[mpix];
    v2f b_s, b_c;
    if (half == 0) { b_s = (v2f){xm, ym};   b_c = (v2f){cm.x, cm.y}; }
    else           { b_s = (v2f){0.f, 0.f}; b_c = (v2f){cm.z, 0.f};  }

    v8f zero = {0.f, 0.f, 0.f, 0.f, 0.f, 0.f, 0.f, 0.f};
    v8f Ss = __builtin_amdgcn_wmma_f32_16x16x4_f32(false, a_s, false, b_s,
                                                   (short)0, zero, false, false);
    v8f Sc = __builtin_amdgcn_wmma_f32_16x16x4_f32(false, a_c, false, b_c,
                                                   (short)0, zero, false, false);

    float sqs_m = xm * xm + ym * ym;
    float sqc_m = cm.w;
    float4 pm = sP[mpix];
#pragma unroll
    for (int r = 0; r < 8; ++r) {
      float ds = sqs_n[r] + sqs_m - 2.0f * Ss[r];
      float dc = sqc_n[r] + sqc_m - 2.0f * Sc[r];
      float w = __expf(-0.5f * ds) * (0.8f * __expf(-0.5f * dc) + 0.2f);
      acc[r].x = fmaf(w, pm.x, acc[r].x);
      acc[r].y = fmaf(w, pm.y, acc[r].y);
      acc[r].z = fmaf(w, pm.z, acc[r].z);
      acc[r].w = fmaf(w, pm.w, acc[r].w);
    }
  }

  // Reduce over the 16 m-columns (lanes within each half-wave).
#pragma unroll
  for (int r = 0; r < 8; ++r) {
#pragma unroll
    for (int off = 1; off < 16; off <<= 1) {
      acc[r].x += __shfl_xor(acc[r].x, off, 32);
      acc[r].y += __shfl_xor(acc[r].y, off, 32);
      acc[r].z += __shfl_xor(acc[r].z, off, 32);
      acc[r].w += __shfl_xor(acc[r].w, off, 32);
    }
  }
  if (col == 0) {
#pragma unroll
    for (int r = 0; r < 8; ++r) comb[nbase + r + 8 * half] = acc[r];
  }
}

// ---------------------------------------------------------------------------
// Update: Potts 3x3 (edge-clamped) + softmax(logits - upd)
//   upd[c] = sum_{q in 3x3} (sum_{c'} comb[c',q]) - comb[c,q]
// ---------------------------------------------------------------------------
__global__ void crf_update(const float* __restrict__ logits,
                           const float4* __restrict__ comb,
                           float4* __restrict__ probs,
                           float* __restrict__ out,
                           int last) {
  int n = blockIdx.x * blockDim.x + threadIdx.x;
  if (n >= Npix) return;
  int i = n / Wd, j = n % Wd;
  float4 vs = make_float4(0.f, 0.f, 0.f, 0.f);
#pragma unroll
  for (int di = -1; di <= 1; ++di) {
    int qi = i + di; qi = qi < 0 ? 0 : (qi > Hh - 1 ? Hh - 1 : qi);
#pragma unroll
    for (int dj = -1; dj <= 1; ++dj) {
      int qj = j + dj; qj = qj < 0 ? 0 : (qj > Wd - 1 ? Wd - 1 : qj);
      float4 q = comb[qi * Wd + qj];
      vs.x += q.x; vs.y += q.y; vs.z += q.z; vs.w += q.w;
    }
  }
  float T = vs.x + vs.y + vs.z + vs.w;
  float l0 = logits[0 * Npix + n] - (T - vs.x);
  float l1 = logits[1 * Npix + n] - (T - vs.y);
  float l2 = logits[2 * Npix + n] - (T - vs.z);
  float l3 = logits[3 * Npix + n] - (T - vs.w);
  float m = fmaxf(fmaxf(l0, l1), fmaxf(l2, l3));
  float e0 = __expf(l0 - m), e1 = __expf(l1 - m);
  float e2 = __expf(l2 - m), e3 = __expf(l3 - m);
  float inv = 1.0f / (e0 + e1 + e2 + e3);
  if (last) {
    out[0 * Npix + n] = e0 * inv;
    out[1 * Npix + n] = e1 * inv;
    out[2 * Npix + n] = e2 * inv;
    out[3 * Npix + n] = e3 * inv;
  } else {
    probs[n] = make_float4(e0 * inv, e1 * inv, e2 * inv, e3 * inv);
  }
}

// ---------------------------------------------------------------------------
extern "C" void kernel_launch(void* const* d_in, const int* in_sizes, int n_in,
                              void* d_out, int out_size, void* d_ws, size_t ws_size,
                              hipStream_t stream) {
  const float* logits = (const float*)d_in[0];  // [1,4,96,96] f32
  const float* ref    = (const float*)d_in[1];  // [1,3,96,96] f32
  float4* probs = (float4*)d_ws;                //  [Npix]
  float4* cfeat = probs + Npix;                 //  [Npix]
  float4* comb  = cfeat + Npix;                 //  [Npix]   (total ws: 432 KB)
  float* out = (float*)d_out;

  crf_init<<<(Npix + 255) / 256, 256, 0, stream>>>(logits, ref, probs, cfeat);

  const size_t lds_bytes = (size_t)Npix * sizeof(float4) * 2;  // 294912 B < 320 KB/WGP
  for (int it = 0; it < 5; ++it) {
    crf_blur<<<NBLK_BLUR, BLK, lds_bytes, stream>>>(probs, cfeat, comb);
    crf_update<<<(Npix + 255) / 256, 256, 0, stream>>>(logits, comb, probs, out, it == 4);
  }
}